// EmbedWeighted_11630771438334
// MI455X (gfx1250) — compile-verified
//
#include <hip/hip_runtime.h>
#include <math.h>

// C[M,N] = A[M,K] * B[K,N], fp32, via V_WMMA_F32_16X16X4_F32 (CDNA5 / gfx1250).
// The reference's gather-then-weighted-sum is exactly this dense GEMM
// (zero weights annihilate the row-0 gathered embedding).

typedef __attribute__((ext_vector_type(2))) float v2f;
typedef __attribute__((ext_vector_type(8))) float v8f;

// 8-arg form: (neg_a, A, neg_b, B, c_mod, C, reuse_a, reuse_b)
#define WMMA_F32_16x16x4(a, b, c) \
  __builtin_amdgcn_wmma_f32_16x16x4_f32(false, (a), false, (b), (short)0, (c), false, false)

// One wave32 per workgroup; each wave owns one 16x16 C tile.
// blockIdx.x -> M tile, blockIdx.y -> N tile. Single-wave workgroups spread
// across all WGPs independent of the WGP count (no barrier allocated).
// NT > 0: compile-time N (folds all B-load offsets into immediates).
template <int NT>
__global__ __launch_bounds__(32) void EmbedWeighted_gemm_wmma(
    const float* __restrict__ A,   // (M, K) multi-hot weights
    const float* __restrict__ B,   // (K, N) embedding table
    float* __restrict__ C,         // (M, N) output
    int M, int K, int N_) {
  const int N     = (NT > 0) ? NT : N_;
  const int lane  = threadIdx.x & 31;
  const int l16   = lane & 15;
  const int khalf = (lane >> 4) << 1;   // 0 (lanes 0-15) or 2 (lanes 16-31)
  const int m0    = blockIdx.x << 4;
  const int n0    = blockIdx.y << 4;

  // A operand (16x4 fp32): lane holds row m = l16, K pair {khalf,khalf+1} -> one b64
  const float* ap = A + (size_t)(m0 + l16) * K + khalf;
  // B operand (4x16 fp32): lane holds col n = l16, K rows {khalf,khalf+1} -> two b32
  const float* bp = B + (size_t)khalf * N + (n0 + l16);

  v8f acc = {};  // 16x16 fp32 accumulator (8 VGPRs)

  int k0 = 0;
  // Main loop: 8 WMMAs / 24 loads in flight per iteration (K step 32).
  for (; k0 + 32 <= K; k0 += 32) {
    __builtin_prefetch(ap + 128, 0, 3);  // pull the A stream ahead
#pragma unroll
    for (int kk = 0; kk < 32; kk += 4) {
      v2f a = *(const v2f*)(ap + kk);
      v2f b;
      b.x = bp[kk * N];        // constant immediate offsets when NT > 0
      b.y = bp[kk * N + N];
      acc = WMMA_F32_16x16x4(a, b, acc);
    }
    ap += 32;
    bp += (size_t)32 * N;
  }
  // Tail: K multiple of 4 (2000 % 32 == 16 -> 4 iterations).
  for (; k0 < K; k0 += 4) {
    v2f a = *(const v2f*)ap;
    v2f b;
    b.x = bp[0];
    b.y = bp[N];
    acc = WMMA_F32_16x16x4(a, b, acc);
    ap += 4;
    bp += (size_t)4 * N;
  }

  // C/D layout: VGPR r <-> row (r + 8*(lane>>4)), col l16.
  float* crow = C + (size_t)(m0 + ((lane >> 4) << 3)) * N + (n0 + l16);
#pragma unroll
  for (int r = 0; r < 8; ++r) {
    crow[(size_t)r * N] = acc[r];
  }
}

extern "C" void kernel_launch(void* const* d_in, const int* in_sizes, int n_in,
                              void* d_out, int out_size, void* d_ws, size_t ws_size,
                              hipStream_t stream) {
  (void)n_in; (void)d_ws; (void)ws_size;
  const float* inputs = (const float*)d_in[0];  // (Bt, V)
  const float* emb    = (const float*)d_in[1];  // (V, D)
  float* out          = (float*)d_out;          // (Bt, D)

  // Recover dims: s0 = Bt*V, s1 = V*D, out_size = Bt*D  =>  D = sqrt(s1*out/s0)
  const double s0 = (double)in_sizes[0];
  const double s1 = (double)in_sizes[1];
  const double so = (double)out_size;
  const int D  = (int)llround(sqrt(s1 * so / s0));
  const int V  = (int)llround(s1 / D);
  const int Bt = (int)llround(so / D);

  // Bt=2048, V=2000, D=64 for this problem; all tile-divisible.
  dim3 grid(Bt / 16, D / 16, 1);
  dim3 block(32, 1, 1);
  if (D == 64) {
    EmbedWeighted_gemm_wmma<64><<<grid, block, 0, stream>>>(inputs, emb, out, Bt, V, D);
  } else {
    EmbedWeighted_gemm_wmma<0><<<grid, block, 0, stream>>>(inputs, emb, out, Bt, V, D);
  }
}